// SamplerHead_12841952215507
// MI455X (gfx1250) — compile-verified
//
#include <hip/hip_runtime.h>

// PointNet++-style SA layer for MI455X (gfx1250, wave32).
// k1: FPS with the whole batch point-cloud resident in 320KB CDNA5 LDS.
// k2: wave-per-keypoint ballot/popcount ball query (both radii, early exit).
// k3: grouped MLP via V_WMMA_F32_16X16X4_F32 (layer1: one WMMA, layer2: 4 chained).

#define DEVFN __device__ __forceinline__

constexpr int   kB   = 4;
constexpr int   kN   = 16384;
constexpr int   kNK  = 2048;
constexpr float kR0  = 0.4f;
constexpr float kR1  = 0.8f;
constexpr int   kNS0 = 16;
constexpr int   kNS1 = 32;

typedef float v2f __attribute__((ext_vector_type(2)));
typedef float v8f __attribute__((ext_vector_type(8)));

DEVFN v8f wmma_f32_16x16x4(v2f a, v2f b, v8f c) {
  // (neg_a, A, neg_b, B, c_mod, C, reuse_a, reuse_b)
  return __builtin_amdgcn_wmma_f32_16x16x4_f32(false, a, false, b, (short)0, c,
                                               false, false);
}

// ---------------------------------------------------------------- kernel 1
// Farthest point sampling. One block per batch; xyz (192KB) + running min
// distance (64KB) live in LDS. 2047 sequential steps of update + argmax.
__global__ __launch_bounds__(1024) void fps_kernel(const float* __restrict__ pts,
                                                   float* __restrict__ coords) {
  __shared__ float sx[kN];
  __shared__ float sy[kN];
  __shared__ float sz[kN];
  __shared__ float sd[kN];
  __shared__ float rv[32];
  __shared__ int   ri[32];
  __shared__ int   scur;

  const int b    = blockIdx.x;
  const int tid  = threadIdx.x;
  const int lane = tid & 31;
  const int wid  = tid >> 5;

  for (int p = tid; p < kN; p += 1024) {
    const float* q = pts + (size_t)(b * kN + p) * 5;
    sx[p] = q[1];
    sy[p] = q[2];
    sz[p] = q[3];
    sd[p] = 1e10f;
  }
  if (tid == 0) scur = 0;
  __syncthreads();

  if (tid == 0) {
    float* c = coords + (size_t)(b * kNK) * 4;
    c[0] = (float)b; c[1] = sx[0]; c[2] = sy[0]; c[3] = sz[0];
  }

  int cur = 0;
  for (int k = 1; k < kNK; ++k) {
    const float kx = sx[cur], ky = sy[cur], kz = sz[cur];
    float bm = -1.0f;
    int   bi = 0;
#pragma unroll
    for (int i = 0; i < 16; ++i) {
      const int   p  = tid + i * 1024;
      const float dx = sx[p] - kx;
      const float dy = sy[p] - ky;
      const float dz = sz[p] - kz;
      const float d  = dx * dx + dy * dy + dz * dz;
      const float nd = fminf(sd[p], d);
      sd[p] = nd;
      if (nd > bm) { bm = nd; bi = p; }
    }
    // wave32 argmax (ties -> smallest index, matching jnp.argmax)
#pragma unroll
    for (int off = 16; off > 0; off >>= 1) {
      const float ov = __shfl_xor(bm, off);
      const int   oi = __shfl_xor(bi, off);
      if (ov > bm || (ov == bm && oi < bi)) { bm = ov; bi = oi; }
    }
    if (lane == 0) { rv[wid] = bm; ri[wid] = bi; }
    __syncthreads();
    if (wid == 0) {
      bm = rv[lane];
      bi = ri[lane];
#pragma unroll
      for (int off = 16; off > 0; off >>= 1) {
        const float ov = __shfl_xor(bm, off);
        const int   oi = __shfl_xor(bi, off);
        if (ov > bm || (ov == bm && oi < bi)) { bm = ov; bi = oi; }
      }
      if (lane == 0) {
        scur = bi;
        float* c = coords + (size_t)(b * kNK + k) * 4;
        c[0] = (float)b; c[1] = sx[bi]; c[2] = sy[bi]; c[3] = sz[bi];
      }
    }
    __syncthreads();
    cur = scur;
  }
}

// ---------------------------------------------------------------- kernel 2
// Ball query: one wave per keypoint; first-16 (r0) and first-32 (r1) point
// indices in ascending order via ballot + popcount prefix append.
__global__ __launch_bounds__(256) void ballq_kernel(const float* __restrict__ pts,
                                                    const float* __restrict__ coords,
                                                    int* __restrict__ idx0,
                                                    int* __restrict__ idx1) {
  const int lane = threadIdx.x & 31;
  const int wid  = threadIdx.x >> 5;
  const int kp   = blockIdx.x * 8 + wid;
  const int b    = kp / kNK;

  const float* c  = coords + (size_t)kp * 4;
  const float  kx = c[1], ky = c[2], kz = c[3];
  const float* base = pts + (size_t)b * kN * 5;

  int c0 = 0, c1 = 0;
  int first0 = -1, first1 = -1;
  const unsigned lt = (1u << lane) - 1u;

  for (int chunk = 0; chunk < kN; chunk += 32) {
    const float* q  = base + (size_t)(chunk + lane) * 5;
    const float  dx = q[1] - kx;
    const float  dy = q[2] - ky;
    const float  dz = q[3] - kz;
    const float  d2 = dx * dx + dy * dy + dz * dz;
    const bool in1 = d2 < kR1 * kR1;
    const bool in0 = d2 < kR0 * kR0;
    const unsigned m1 = (unsigned)__ballot(in1);
    const unsigned m0 = (unsigned)__ballot(in0);
    if (c1 == 0 && m1) first1 = chunk + __ffs(m1) - 1;
    if (c0 == 0 && m0) first0 = chunk + __ffs(m0) - 1;
    if (in1) {
      const int pos = c1 + __popc(m1 & lt);
      if (pos < kNS1) idx1[(size_t)kp * kNS1 + pos] = chunk + lane;
    }
    if (in0) {
      const int pos = c0 + __popc(m0 & lt);
      if (pos < kNS0) idx0[(size_t)kp * kNS0 + pos] = chunk + lane;
    }
    c1 = min(kNS1, c1 + __popc(m1));
    c0 = min(kNS0, c0 + __popc(m0));
    if (c1 >= kNS1 && c0 >= kNS0) break;
  }
  // pad with first index (or -1 => "empty", rows zeroed downstream)
  if (lane >= c1) idx1[(size_t)kp * kNS1 + lane] = first1;
  if (lane < kNS0 && lane >= c0) idx0[(size_t)kp * kNS0 + lane] = first0;
}

// ---------------------------------------------------------------- kernel 3
// Grouped 2-layer MLP (scale+bias+relu) + maxpool, one wave per keypoint.
// Layer1: D(16x16) = A(16 samples x 4 ch) * B(4 x 16 out) via one
// v_wmma_f32_16x16x4_f32. Layer2: K=16 as 4 chained K=4 WMMAs; the
// D-layout -> A-layout relayout bounces through a padded LDS tile.
template <int NTILE>
DEVFN void radius_pass(const int* __restrict__ idxlist, int kp, int ns,
                       const float* __restrict__ pbase,
                       float kx, float ky, float kz,
                       const float* __restrict__ w0, const float* __restrict__ g0,
                       const float* __restrict__ b0,
                       const float* __restrict__ w1, const float* __restrict__ g1,
                       const float* __restrict__ b1,
                       float (*T)[17], int lane, float* __restrict__ out) {
  const int s  = lane & 15;   // sample row within tile / out-channel column
  const int hf = lane >> 4;   // lane half selects K (A/B) or M block (D)
  const int n  = s;

  // B matrix (w0^T, 4x16): lane half hf holds K = 2*hf, 2*hf+1 of column n
  v2f bw0;
  bw0.x = w0[n * 4 + 2 * hf + 0];
  bw0.y = w0[n * 4 + 2 * hf + 1];
  // Layer-2 B chunks (w1^T, 16x16 split into 4 K-slabs of 4)
  v2f bw1[4];
#pragma unroll
  for (int kc = 0; kc < 4; ++kc) {
    bw1[kc].x = w1[n * 16 + kc * 4 + 2 * hf + 0];
    bw1[kc].y = w1[n * 16 + kc * 4 + 2 * hf + 1];
  }
  const float gs0 = g0[n], bs0 = b0[n];
  const float gs1 = g1[n], bs1 = b1[n];

  float pool = 0.0f;
#pragma unroll
  for (int t = 0; t < NTILE; ++t) {
    // Gather group row (dx,dy,dz,intensity); branchless so EXEC stays full
    // for WMMA. iv < 0 (empty group) zeroes the row, matching the reference.
    const int   iv  = idxlist[(size_t)kp * ns + t * 16 + s];
    const int   ivc = iv < 0 ? 0 : iv;
    const float msk = iv < 0 ? 0.0f : 1.0f;
    const float* q  = pbase + (size_t)ivc * 5;
    const float e0 = (q[1] - kx) * msk;
    const float e1 = (q[2] - ky) * msk;
    const float e2 = (q[3] - kz) * msk;
    const float e3 = q[4] * msk;

    v2f a;
    a.x = hf ? e2 : e0;
    a.y = hf ? e3 : e1;
    v8f acc = {0.f, 0.f, 0.f, 0.f, 0.f, 0.f, 0.f, 0.f};
    acc = wmma_f32_16x16x4(a, bw0, acc);

    // h = relu(D * g0 + b0); stage into padded LDS tile for relayout
#pragma unroll
    for (int r = 0; r < 8; ++r) {
      const float h = acc[r] * gs0 + bs0;
      T[r + 8 * hf][n] = h > 0.f ? h : 0.f;
    }
    __syncthreads();

    v8f acc2 = {0.f, 0.f, 0.f, 0.f, 0.f, 0.f, 0.f, 0.f};
#pragma unroll
    for (int kc = 0; kc < 4; ++kc) {
      v2f a2;
      a2.x = T[s][kc * 4 + 2 * hf + 0];
      a2.y = T[s][kc * 4 + 2 * hf + 1];
      acc2 = wmma_f32_16x16x4(a2, bw1[kc], acc2);
    }
    __syncthreads();

#pragma unroll
    for (int r = 0; r < 8; ++r) {
      float h = acc2[r] * gs1 + bs1;
      h = h > 0.f ? h : 0.f;
      pool = fmaxf(pool, h);
    }
  }
  // combine the two M halves, lanes 0-15 hold the pooled channel values
  const float other = __shfl_xor(pool, 16);
  pool = fmaxf(pool, other);
  if (hf == 0) out[n] = pool;
}

__global__ __launch_bounds__(256) void mlp_kernel(
    const float* __restrict__ pts, const float* __restrict__ coords,
    const int* __restrict__ idx0, const int* __restrict__ idx1,
    const float* __restrict__ w0r0, const float* __restrict__ g0r0,
    const float* __restrict__ b0r0, const float* __restrict__ w1r0,
    const float* __restrict__ g1r0, const float* __restrict__ b1r0,
    const float* __restrict__ w0r1, const float* __restrict__ g0r1,
    const float* __restrict__ b0r1, const float* __restrict__ w1r1,
    const float* __restrict__ g1r1, const float* __restrict__ b1r1,
    float* __restrict__ feat) {
  __shared__ float Tall[8][16][17];  // one padded 16x16 tile per wave
  const int lane = threadIdx.x & 31;
  const int wid  = threadIdx.x >> 5;
  const int kp   = blockIdx.x * 8 + wid;
  const int b    = kp / kNK;

  const float* c  = coords + (size_t)kp * 4;
  const float  kx = c[1], ky = c[2], kz = c[3];
  const float* pbase = pts + (size_t)b * kN * 5;
  float* out = feat + (size_t)kp * 32;

  radius_pass<1>(idx0, kp, kNS0, pbase, kx, ky, kz,
                 w0r0, g0r0, b0r0, w1r0, g1r0, b1r0, Tall[wid], lane, out);
  radius_pass<2>(idx1, kp, kNS1, pbase, kx, ky, kz,
                 w0r1, g0r1, b0r1, w1r1, g1r1, b1r1, Tall[wid], lane, out + 16);
}

// ---------------------------------------------------------------- launcher
extern "C" void kernel_launch(void* const* d_in, const int* in_sizes, int n_in,
                              void* d_out, int out_size, void* d_ws, size_t ws_size,
                              hipStream_t stream) {
  (void)in_sizes; (void)n_in; (void)out_size; (void)ws_size;
  const float* pts  = (const float*)d_in[0];
  const float* w0r0 = (const float*)d_in[1];
  const float* g0r0 = (const float*)d_in[2];
  const float* b0r0 = (const float*)d_in[3];
  const float* w1r0 = (const float*)d_in[4];
  const float* g1r0 = (const float*)d_in[5];
  const float* b1r0 = (const float*)d_in[6];
  const float* w0r1 = (const float*)d_in[7];
  const float* g0r1 = (const float*)d_in[8];
  const float* b0r1 = (const float*)d_in[9];
  const float* w1r1 = (const float*)d_in[10];
  const float* g1r1 = (const float*)d_in[11];
  const float* b1r1 = (const float*)d_in[12];

  float* feat   = (float*)d_out;                      // (B*NK, 32)
  float* coords = feat + (size_t)kB * kNK * 32;       // (B*NK, 4)

  int* idx1 = (int*)d_ws;                             // B*NK*32 ints
  int* idx0 = idx1 + (size_t)kB * kNK * kNS1;         // B*NK*16 ints

  fps_kernel<<<kB, 1024, 0, stream>>>(pts, coords);
  ballq_kernel<<<(kB * kNK) / 8, 256, 0, stream>>>(pts, coords, idx0, idx1);
  mlp_kernel<<<(kB * kNK) / 8, 256, 0, stream>>>(
      pts, coords, idx0, idx1,
      w0r0, g0r0, b0r0, w1r0, g1r0, b1r0,
      w0r1, g0r1, b0r1, w1r1, g1r1, b1r1, feat);
}